// FocalLoss_36163624632413
// MI455X (gfx1250) — compile-verified
//
#include <hip/hip_runtime.h>
#include <hip/hip_bf16.h>

#define B_IMG 8
#define A_N   100000
#define K_CLS 80
#define M_ANN 32

typedef __attribute__((ext_vector_type(2))) float v2f;
typedef __attribute__((ext_vector_type(4))) float f4;
typedef __attribute__((ext_vector_type(8))) float v8f;

// Wave32 sum reduction using the matrix pipe:
// B-operand (4x16) holds one partial per lane (b.y = 0), A = ones(16x4).
// D[m][n] = sum_k B[k][n]  -> D vgpr0, lane L = s[L&15] + s[(L&15)+16].
// 4 xor-shuffles then sum the 16 distinct columns. All lanes return the total.
__device__ __forceinline__ float wave_reduce_sum(float s) {
    v2f a; a.x = 1.0f; a.y = 1.0f;
    v2f b; b.x = s;    b.y = 0.0f;
    v8f c = {};
    c = __builtin_amdgcn_wmma_f32_16x16x4_f32(false, a, false, b, (short)0, c,
                                              false, false);
    float v = c[0];
    v += __shfl_xor(v, 1, 32);
    v += __shfl_xor(v, 2, 32);
    v += __shfl_xor(v, 4, 32);
    v += __shfl_xor(v, 8, 32);
    return v;
}

__device__ __forceinline__ float smooth_l1(float d) {
    d = fabsf(d);
    return (d <= (1.0f / 9.0f)) ? 4.5f * d * d : d - (0.5f / 9.0f);
}

__global__ void init_kernel(float* acc) {
    int i = threadIdx.x;
    if (i < 3 * B_IMG) acc[i] = 0.0f;
}

// Per-anchor: IoU max/argmax vs 32 annotations (in LDS), emit label-or-neg,
// accumulate num_pos and smooth-L1 regression sum per image.
__global__ void __launch_bounds__(256)
anchor_kernel(const float* __restrict__ anchors,
              const float* __restrict__ regressions,
              const float* __restrict__ annotations,
              int*   __restrict__ labels,
              float* __restrict__ reg_sum,
              float* __restrict__ num_pos) {
    __shared__ float ann[M_ANN * 5];
    __shared__ float redA[8], redB[8];
    const int b = blockIdx.y;
    const int a = blockIdx.x * blockDim.x + threadIdx.x;

    if (threadIdx.x < M_ANN * 5)
        ann[threadIdx.x] = annotations[b * M_ANN * 5 + threadIdx.x];
    __syncthreads();

    float my_reg = 0.0f, my_np = 0.0f;
    if (a < A_N) {
        f4 ab = ((const f4*)anchors)[a];
        float ax1 = ab.x, ay1 = ab.y, ax2 = ab.z, ay2 = ab.w;
        float best = -2.0f; int bj = 0;
        #pragma unroll 4
        for (int j = 0; j < M_ANN; ++j) {
            float bx1 = ann[j * 5 + 0], by1 = ann[j * 5 + 1];
            float bx2 = ann[j * 5 + 2], by2 = ann[j * 5 + 3];
            float lb  = ann[j * 5 + 4];
            float iw = fmaxf(fminf(ax2, bx2) - fmaxf(ax1, bx1), 0.0f);
            float ih = fmaxf(fminf(ay2, by2) - fmaxf(ay1, by1), 0.0f);
            float inter = iw * ih;
            float ua = (ax2 - ax1) * (ay2 - ay1) +
                       (bx2 - bx1) * (by2 - by1) - inter;
            float iou = inter / fmaxf(ua, 1e-8f);
            iou = (lb != -1.0f) ? iou : -1.0f;   // mask invalid annotations
            if (iou > best) { best = iou; bj = j; }  // first-max semantics
        }
        bool positive = best >= 0.5f;
        labels[(size_t)b * A_N + a] = positive ? (int)ann[bj * 5 + 4] : -1;
        if (positive) {
            my_np = 1.0f;
            float gx1 = ann[bj * 5 + 0], gy1 = ann[bj * 5 + 1];
            float gx2 = ann[bj * 5 + 2], gy2 = ann[bj * 5 + 3];
            float aw = ax2 - ax1, ah = ay2 - ay1;
            float acx = ax1 + 0.5f * aw, acy = ay1 + 0.5f * ah;
            float gw = gx2 - gx1, gh = gy2 - gy1;
            float gcx = gx1 + 0.5f * gw, gcy = gy1 + 0.5f * gh;  // before clip
            gw = fmaxf(gw, 1.0f); gh = fmaxf(gh, 1.0f);
            float t0 = ((gcx - acx) / aw) / 0.1f;
            float t1 = ((gcy - acy) / ah) / 0.1f;
            float t2 = __logf(gw / aw) / 0.2f;
            float t3 = __logf(gh / ah) / 0.2f;
            f4 r = *(const f4*)(regressions + ((size_t)b * A_N + a) * 4);
            my_reg = smooth_l1(t0 - r.x) + smooth_l1(t1 - r.y) +
                     smooth_l1(t2 - r.z) + smooth_l1(t3 - r.w);
        }
    }
    // uniform reduction (EXEC all-ones for WMMA)
    float wr = wave_reduce_sum(my_reg);
    float wn = wave_reduce_sum(my_np);
    int wave = threadIdx.x >> 5, lane = threadIdx.x & 31;
    if (lane == 0) { redA[wave] = wr; redB[wave] = wn; }
    __syncthreads();
    if (threadIdx.x == 0) {
        float sr = 0.0f, sn = 0.0f;
        #pragma unroll
        for (int i = 0; i < 8; ++i) { sr += redA[i]; sn += redB[i]; }
        atomicAdd(&reg_sum[b], sr);
        atomicAdd(&num_pos[b], sn);
    }
}

// Stream 256 MB of classifications once. Each thread owns whole anchors:
// one label load per 20 b128 loads, no index division, fully unrolled inner
// loop so all 20 non-temporal loads issue back-to-back (latency hiding).
// Per-image sum via WMMA wave reduce + one atomic per block.
__global__ void __launch_bounds__(256)
cls_kernel(const float* __restrict__ cls,
           const int*   __restrict__ labels,
           float* __restrict__ cls_sum) {
    __shared__ float red[8];
    const int b = blockIdx.y;
    const f4* cls4 = (const f4*)(cls + (size_t)b * A_N * K_CLS);
    const int* lab = labels + (size_t)b * A_N;
    const int stride = gridDim.x * blockDim.x;   // threads per image

    float acc = 0.0f;
    for (int a = blockIdx.x * blockDim.x + threadIdx.x; a < A_N; a += stride) {
        const int lbl = lab[a];                  // one label per anchor
        const bool pos = lbl >= 0;
        const f4* base = cls4 + (size_t)a * (K_CLS / 4);
        #pragma unroll
        for (int j = 0; j < K_CLS / 4; ++j) {
            f4 v = __builtin_nontemporal_load(base + j);
            #pragma unroll
            for (int t = 0; t < 4; ++t) {
                float c = fminf(fmaxf(v[t], 1e-4f), 1.0f - 1e-4f);
                float e = __expf(-c);
                float p = 1.0f / (1.0f + e);
                float lp = __logf(p);            // softplus(c) = c - lp
                bool sel = pos && ((j * 4 + t) != lbl);  // positive, k != label
                float q = sel ? p : (1.0f - p);
                float r = sel ? 0.75f * (c - lp) : (-0.25f) * lp;
                acc = fmaf(q * q, r, acc);
            }
        }
    }
    float w = wave_reduce_sum(acc);              // uniform: EXEC all-ones
    int wave = threadIdx.x >> 5, lane = threadIdx.x & 31;
    if (lane == 0) red[wave] = w;
    __syncthreads();
    if (threadIdx.x == 0) {
        float s = 0.0f;
        #pragma unroll
        for (int i = 0; i < 8; ++i) s += red[i];
        atomicAdd(&cls_sum[b], s);
    }
}

__global__ void finalize_kernel(const float* __restrict__ cls_sum,
                                const float* __restrict__ reg_sum,
                                const float* __restrict__ num_pos,
                                const float* __restrict__ annotations,
                                float* __restrict__ out) {
    if (threadIdx.x == 0) {
        float ct = 0.0f, rt = 0.0f;
        for (int b = 0; b < B_IMG; ++b) {
            bool has = false;
            for (int j = 0; j < M_ANN; ++j)
                if (annotations[(b * M_ANN + j) * 5 + 4] != -1.0f) has = true;
            float np = num_pos[b];
            float cl = cls_sum[b] / fmaxf(np, 0.01f);
            float rl = (np > 0.0f) ? reg_sum[b] / fmaxf(np * 4.0f, 1.0f) : 0.0f;
            ct += has ? cl : 0.0f;
            rt += has ? rl : 0.0f;
        }
        out[0] = ct / (float)B_IMG;
        out[1] = rt / (float)B_IMG;
    }
}

extern "C" void kernel_launch(void* const* d_in, const int* in_sizes, int n_in,
                              void* d_out, int out_size, void* d_ws, size_t ws_size,
                              hipStream_t stream) {
    const float* cls  = (const float*)d_in[0];  // (B, A, K) f32
    const float* reg  = (const float*)d_in[1];  // (B, A, 4) f32
    const float* anch = (const float*)d_in[2];  // (1, A, 4) f32
    const float* ann  = (const float*)d_in[3];  // (B, M, 5) f32
    float* out = (float*)d_out;                 // [mean_cls, mean_reg]

    float* acc    = (float*)d_ws;               // [0,B)=cls [B,2B)=reg [2B,3B)=npos
    int*   labels = (int*)((char*)d_ws + 256);  // B*A ints (3.2 MB)

    init_kernel<<<1, 64, 0, stream>>>(acc);

    dim3 g1((A_N + 255) / 256, B_IMG);
    anchor_kernel<<<g1, 256, 0, stream>>>(anch, reg, ann, labels,
                                          acc + B_IMG, acc + 2 * B_IMG);

    dim3 g2(128, B_IMG);                        // 32768 threads/image, ~3 anchors ea.
    cls_kernel<<<g2, 256, 0, stream>>>(cls, labels, acc);

    finalize_kernel<<<1, 32, 0, stream>>>(acc, acc + B_IMG, acc + 2 * B_IMG,
                                          ann, out);
}